// AttnPhi_70179765616671
// MI455X (gfx1250) — compile-verified
//
#include <hip/hip_runtime.h>
#include <math.h>

#define NUM_HEADS 8
#define D_ATT     96
#define D_MODEL   768
#define S_TOTAL   4096
#define BATCH     16
#define SPLITS    8
#define S_PER     (S_TOTAL / SPLITS)   // 512 positions per split-WG
#define TILE_P    128                  // positions staged in LDS per tile
#define NTILES    (S_PER / TILE_P)     // 4
// TDM-padded LDS tile layout: 2 dwords of pad after every 32 dwords streamed.
// element (row,d) lives at dword index row*102 + d + 2*(d>>5); rows are 96 dw.
#define TIDX(row, d) ((row) * 102 + (d) + 2 * ((d) >> 5))
#define TILE_DW   (TILE_P * 102)       // 13056 dwords = 52224 B

typedef __attribute__((ext_vector_type(2))) float v2f;
typedef __attribute__((ext_vector_type(8))) float v8f;
typedef __attribute__((ext_vector_type(4))) unsigned int v4u;
typedef __attribute__((ext_vector_type(8))) int v8i;
typedef __attribute__((ext_vector_type(4))) int v4i;

__device__ __forceinline__ float waveMax(float v) {
#pragma unroll
  for (int o = 16; o > 0; o >>= 1) v = fmaxf(v, __shfl_xor(v, o, 32));
  return v;
}
__device__ __forceinline__ float waveSum(float v) {
#pragma unroll
  for (int o = 16; o > 0; o >>= 1) v += __shfl_xor(v, o, 32);
  return v;
}

// ---------------------------------------------------------------------------
// TDM: DMA one 2D tile (TILE_P rows x 96 f32, row stride 768 f32) from global
// memory into LDS at byte offset lds_off, with 2-dword pad every 32 dwords.
// Descriptor packing per CDNA5 ISA 8.3-8.5. Issue from one wave; EXEC ignored.
// ---------------------------------------------------------------------------
__device__ __forceinline__ void tdm_load_tile(const float* gsrc, unsigned lds_off) {
  unsigned long long ga = (unsigned long long)(uintptr_t)gsrc;
  v4u g0;
  g0[0] = 1u;                                  // count=1 valid descriptor
  g0[1] = lds_off;                             // lds_addr (bytes)
  g0[2] = (unsigned)ga;                        // global_addr[31:0]
  g0[3] = (unsigned)(ga >> 32) | (2u << 30);   // global_addr[56:32] | type=2
  v8i g1;
  g1[0] = (2 << 16)      // data_size = 4 bytes
        | (1 << 20)      // pad_enable
        | (4 << 22)      // pad_interval code 4 = 32 dwords
        | (1 << 25);     // pad_amount  code 1 = 2 dwords
  g1[1] = 0;                                   // atomic_barrier addr / td0[15:0]=0
  g1[2] = 0x4000;                              // tensor_dim0 = 1<<30 (hi), td1 lo = 0
  g1[3] = 0x4000 | (D_ATT << 16);              // tensor_dim1 hi | tile_dim0 = 96
  g1[4] = TILE_P;                              // tile_dim1 = 128, tile_dim2 = 0
  g1[5] = D_MODEL;                             // tensor_dim0_stride = 768
  g1[6] = 0;
  g1[7] = 0;
  v4i g2 = {0, 0, 0, 0};
  v4i g3 = {0, 0, 0, 0};
  asm volatile("tensor_load_to_lds %0, %1, %2, %3"
               :: "s"(g0), "s"(g1), "s"(g2), "s"(g3)
               : "memory");
}

// ---------------------------------------------------------------------------
// Kernel 1: fill sinusoidal positional-encoding table pe[h][s][d] (L2-resident)
// ---------------------------------------------------------------------------
__global__ __launch_bounds__(256) void pe_fill(float* __restrict__ pe) {
  int idx = blockIdx.x * 256 + threadIdx.x;
  if (idx >= NUM_HEADS * S_TOTAL * D_ATT) return;
  int d = idx % D_ATT;
  int s = (idx / D_ATT) % S_TOTAL;
  int h = idx / (D_ATT * S_TOTAL);
  int c = h * D_ATT + d;        // channel in [0,768)
  int j = c >> 1;
  const float coef = (float)(-9.210340371976184 / 768.0);   // -ln(1e4)/768
  float dv  = expf(coef * (float)(2 * j));
  float ang = (float)s * dv;
  float v   = (c & 1) ? cosf(ang) : sinf(ang);
  pe[idx] = v * 0.03608439182435161f;                       // * 768^-0.5
}

// ---------------------------------------------------------------------------
// Kernel 2: per-(b,h,split) partial attention.
//   Phase 1: TDM-staged tiles -> WMMA f32 16x16x4 score GEMV.
//   Softmax stats (local max m, local sum l) over this split.
//   Phase 2: weighted sum of (key + pe) -> 96-dim partial accumulator.
// ---------------------------------------------------------------------------
__global__ __launch_bounds__(256) void attn_partial(
    const float* __restrict__ src, const unsigned char* __restrict__ mask,
    const float* __restrict__ query, const float* __restrict__ pe,
    float* __restrict__ pm, float* __restrict__ pl, float* __restrict__ pacc) {
  __shared__ float tile[TILE_DW];       // 52224 B TDM target; reused for partials
  __shared__ float qbuf[D_ATT];
  __shared__ float sc[S_PER];           // scores -> unnormalized weights
  __shared__ float red[8];

  int bh = blockIdx.x / SPLITS;
  int sp = blockIdx.x % SPLITS;
  int b = bh >> 3, h = bh & 7;
  int t = threadIdx.x, lane = t & 31, w = t >> 5;
  const float* srcbh = src + (size_t)b * S_TOTAL * D_MODEL + h * D_ATT;
  const float* pehd  = pe + (size_t)h * S_TOTAL * D_ATT;
  int s0 = sp * S_PER;
  unsigned tile_lds = (unsigned)(uintptr_t)&tile[0];  // low 32 bits = LDS offset

  if (t < D_ATT) qbuf[t] = query[h * D_ATT + t];
  __syncthreads();

  // ---------------- Phase 1: TDM stage + WMMA scores ----------------
  for (int tl = 0; tl < NTILES; ++tl) {
    int sbase = s0 + tl * TILE_P;
    if (w == 0) {
      tdm_load_tile(srcbh + (size_t)sbase * D_MODEL, tile_lds);
      __builtin_amdgcn_s_wait_tensorcnt(0);
    }
    if (tl + 1 < NTILES) {  // hint next tile toward L2 while we compute
      const float* nf = srcbh + (size_t)(sbase + TILE_P + (t & 127)) * D_MODEL + (t >> 7) * 48;
      __builtin_prefetch(nf, 0, 3);
    }
    __syncthreads();

    // Wave w computes scores for rows [w*16, w*16+16).
    // A (16x4 f32): lanes 0-15 hold {K0,K1} of row M=lane; lanes 16-31 hold {K2,K3}.
    // B (4x16 f32): q replicated across all 16 columns -> every column of D = scores.
    int row  = w * 16 + (lane & 15);
    int koff = (lane >> 4) * 2;
    v8f acc = {};
#pragma unroll
    for (int dbase = 0; dbase < D_ATT; dbase += 4) {
      int d0 = dbase + koff;
      v2f a, bb;
      a.x  = tile[TIDX(row, d0)];
      a.y  = tile[TIDX(row, d0) + 1];
      bb.x = qbuf[d0];
      bb.y = qbuf[d0 + 1];
      acc = __builtin_amdgcn_wmma_f32_16x16x4_f32(
          /*neg_a=*/false, a, /*neg_b=*/false, bb,
          /*c_mod=*/(short)0, acc, /*reuse_a=*/false, /*reuse_b=*/false);
    }
    // D layout: VGPR r holds M=r (lanes 0-15) and M=8+r (lanes 16-31); cols identical.
    if ((lane & 15) == 0) {
      int rb = w * 16 + (lane >> 4) * 8;
#pragma unroll
      for (int r = 0; r < 8; ++r) {
        int sl = tl * TILE_P + rb + r;
        float v = acc[r];
        if (mask[b * S_TOTAL + s0 + sl]) v = -INFINITY;
        sc[sl] = v;
      }
    }
    __syncthreads();              // tile consumed; next TDM may overwrite
  }

  // ---------------- Softmax stats for this split ----------------
  float mloc = -INFINITY;
  for (int i = t; i < S_PER; i += 256) mloc = fmaxf(mloc, sc[i]);
  mloc = waveMax(mloc);
  if (lane == 0) red[w] = mloc;
  __syncthreads();
  float M = red[0];
#pragma unroll
  for (int i = 1; i < 8; ++i) M = fmaxf(M, red[i]);
  __syncthreads();

  float lloc = 0.f;
  for (int i = t; i < S_PER; i += 256) {
    float e = (M == -INFINITY) ? 0.f : __expf(sc[i] - M);
    sc[i] = e;                    // unnormalized weight
    lloc += e;
  }
  lloc = waveSum(lloc);
  if (lane == 0) red[w] = lloc;
  __syncthreads();
  float L = 0.f;
#pragma unroll
  for (int i = 0; i < 8; ++i) L += red[i];
  if (t == 0) { pm[bh * SPLITS + sp] = M; pl[bh * SPLITS + sp] = L; }

  // ---------------- Phase 2: pooled partial = sum_s w_s * (key + pe) ----------
  // Lane L owns d in {L, L+32, L+64}; global loads fully coalesced and L2-hot.
  float a0 = 0.f, a1 = 0.f, a2 = 0.f;
  int sw0 = w * (S_PER / 8);      // 64 positions per wave
  for (int i = 0; i < S_PER / 8; ++i) {
    int sl = sw0 + i;
    float wgt = sc[sl];
    const float* kp = srcbh + (size_t)(s0 + sl) * D_MODEL;
    const float* pp = pehd + (size_t)(s0 + sl) * D_ATT;
    a0 = fmaf(wgt, kp[lane]      + pp[lane],      a0);
    a1 = fmaf(wgt, kp[lane + 32] + pp[lane + 32], a1);
    a2 = fmaf(wgt, kp[lane + 64] + pp[lane + 64], a2);
  }
  __syncthreads();                // tile buffer free -> reuse as partials
  float* redv = tile;             // [8][96]
  redv[w * D_ATT + lane]      = a0;
  redv[w * D_ATT + lane + 32] = a1;
  redv[w * D_ATT + lane + 64] = a2;
  __syncthreads();
  if (t < D_ATT) {
    float s = 0.f;
#pragma unroll
    for (int ww = 0; ww < 8; ++ww) s += redv[ww * D_ATT + t];
    pacc[(bh * SPLITS + sp) * D_ATT + t] = s;
  }
}

// ---------------------------------------------------------------------------
// Kernel 3: combine split partials with global max rescale; write output.
// out layout: [B, 768] with b*768 + h*96 + d == bh*96 + d.
// ---------------------------------------------------------------------------
__global__ void attn_combine(const float* __restrict__ pm, const float* __restrict__ pl,
                             const float* __restrict__ pacc, float* __restrict__ out) {
  int bh = blockIdx.x;
  int d = threadIdx.x;
  if (d >= D_ATT) return;
  float M = -INFINITY;
#pragma unroll
  for (int sp = 0; sp < SPLITS; ++sp) M = fmaxf(M, pm[bh * SPLITS + sp]);
  float L = 0.f, s = 0.f;
#pragma unroll
  for (int sp = 0; sp < SPLITS; ++sp) {
    float msp = pm[bh * SPLITS + sp];
    float f = (msp == -INFINITY) ? 0.f : __expf(msp - M);
    L += f * pl[bh * SPLITS + sp];
    s += f * pacc[(bh * SPLITS + sp) * D_ATT + d];
  }
  out[bh * D_ATT + d] = s / L;
}

// ---------------------------------------------------------------------------
// Fallback (workspace too small): single-kernel two-pass, PE computed inline.
// ---------------------------------------------------------------------------
__global__ __launch_bounds__(256) void attn_fallback(
    const float* __restrict__ src, const unsigned char* __restrict__ mask,
    const float* __restrict__ query, float* __restrict__ out) {
  __shared__ float sc[S_TOTAL];
  __shared__ float qbuf[D_ATT];
  __shared__ float red[8];
  __shared__ float redv[8 * D_ATT];
  int bh = blockIdx.x, b = bh >> 3, h = bh & 7;
  int t = threadIdx.x, lane = t & 31, w = t >> 5;
  const float* srcbh = src + (size_t)b * S_TOTAL * D_MODEL + h * D_ATT;
  if (t < D_ATT) qbuf[t] = query[h * D_ATT + t];
  __syncthreads();
  for (int s = t; s < S_TOTAL; s += 256) {
    const float* kp = srcbh + (size_t)s * D_MODEL;
    float acc = 0.f;
    for (int d = 0; d < D_ATT; ++d) acc = fmaf(kp[d], qbuf[d], acc);
    sc[s] = mask[b * S_TOTAL + s] ? -INFINITY : acc;
  }
  __syncthreads();
  float mloc = -INFINITY;
  for (int i = t; i < S_TOTAL; i += 256) mloc = fmaxf(mloc, sc[i]);
  mloc = waveMax(mloc);
  if (lane == 0) red[w] = mloc;
  __syncthreads();
  float M = red[0];
  for (int i = 1; i < 8; ++i) M = fmaxf(M, red[i]);
  __syncthreads();
  float lloc = 0.f;
  for (int i = t; i < S_TOTAL; i += 256) {
    float e = (M == -INFINITY) ? 0.f : __expf(sc[i] - M);
    sc[i] = e; lloc += e;
  }
  lloc = waveSum(lloc);
  if (lane == 0) red[w] = lloc;
  __syncthreads();
  float L = 0.f;
  for (int i = 0; i < 8; ++i) L += red[i];

  const float coef = (float)(-9.210340371976184 / 768.0);
  const float pesc = 0.03608439182435161f;
  int c0 = h * D_ATT + lane;                 // c0, c0+32, c0+64 share parity
  float dv0 = __expf(coef * (float)(2 * (c0 >> 1)));
  float dv1 = __expf(coef * (float)(2 * ((c0 + 32) >> 1)));
  float dv2 = __expf(coef * (float)(2 * ((c0 + 64) >> 1)));
  bool odd = (c0 & 1);
  float a0 = 0.f, a1 = 0.f, a2 = 0.f;
  int sw0 = w * (S_TOTAL / 8);
  for (int i = 0; i < S_TOTAL / 8; ++i) {
    int s = sw0 + i;
    float wgt = sc[s];
    const float* kp = srcbh + (size_t)s * D_MODEL;
    float fs = (float)s;
    float p0 = odd ? cosf(fs * dv0) : sinf(fs * dv0);
    float p1 = odd ? cosf(fs * dv1) : sinf(fs * dv1);
    float p2 = odd ? cosf(fs * dv2) : sinf(fs * dv2);
    a0 = fmaf(wgt, kp[lane]      + p0 * pesc, a0);
    a1 = fmaf(wgt, kp[lane + 32] + p1 * pesc, a1);
    a2 = fmaf(wgt, kp[lane + 64] + p2 * pesc, a2);
  }
  redv[w * D_ATT + lane]      = a0;
  redv[w * D_ATT + lane + 32] = a1;
  redv[w * D_ATT + lane + 64] = a2;
  __syncthreads();
  if (t < D_ATT) {
    float ssum = 0.f;
    for (int ww = 0; ww < 8; ++ww) ssum += redv[ww * D_ATT + t];
    out[bh * D_ATT + t] = ssum / L;
  }
}

// ---------------------------------------------------------------------------
extern "C" void kernel_launch(void* const* d_in, const int* in_sizes, int n_in,
                              void* d_out, int out_size, void* d_ws, size_t ws_size,
                              hipStream_t stream) {
  const float* src = (const float*)d_in[0];
  const unsigned char* mask = (const unsigned char*)d_in[1];
  const float* query = (const float*)d_in[2];
  float* out = (float*)d_out;

  const size_t PE_N = (size_t)NUM_HEADS * S_TOTAL * D_ATT;          // 3,145,728
  const size_t NP   = (size_t)BATCH * NUM_HEADS * SPLITS;           // 1024
  const size_t need = (PE_N + NP + NP + NP * D_ATT) * sizeof(float);

  if (ws_size >= need) {
    float* pe   = (float*)d_ws;
    float* pm   = pe + PE_N;
    float* pl   = pm + NP;
    float* pacc = pl + NP;
    pe_fill<<<(int)((PE_N + 255) / 256), 256, 0, stream>>>(pe);
    attn_partial<<<(int)NP, 256, 0, stream>>>(src, mask, query, pe, pm, pl, pacc);
    attn_combine<<<BATCH * NUM_HEADS, D_ATT, 0, stream>>>(pm, pl, pacc, out);
  } else {
    attn_fallback<<<BATCH * NUM_HEADS, 256, 0, stream>>>(src, mask, query, out);
  }
}